// Attention_89077621719647
// MI455X (gfx1250) — compile-verified
//
#include <hip/hip_runtime.h>
#include <hip/hip_bf16.h>
#include <math.h>

// Problem constants (from reference): B=16, T=2048, D=1024
#define TT 2048
#define DD 1024
#define BB 16
#define TCHUNK 128   // t-rows per block; gridDim.z = TT/TCHUNK = 16

typedef __attribute__((ext_vector_type(2))) float v2f;
typedef __attribute__((ext_vector_type(8))) float v8f;

// ---------------------------------------------------------------------------
// Kernel 1: softmax(weights) -> scores[T] in workspace, and zero d_out.
// One block, trivially cheap vs. the 134MB stream.
// ---------------------------------------------------------------------------
__global__ void attn_prep_kernel(const float* __restrict__ weights,
                                 float* __restrict__ scores,
                                 float* __restrict__ out) {
    __shared__ float red[256];
    const int tid = threadIdx.x;

    // global max
    float m = -INFINITY;
    for (int i = tid; i < TT; i += 256) m = fmaxf(m, weights[i]);
    red[tid] = m;
    __syncthreads();
    for (int s = 128; s > 0; s >>= 1) {
        if (tid < s) red[tid] = fmaxf(red[tid], red[tid + s]);
        __syncthreads();
    }
    const float gmax = red[0];
    __syncthreads();

    // sum of exp
    float sum = 0.0f;
    for (int i = tid; i < TT; i += 256) sum += __expf(weights[i] - gmax);
    red[tid] = sum;
    __syncthreads();
    for (int s = 128; s > 0; s >>= 1) {
        if (tid < s) red[tid] += red[tid + s];
        __syncthreads();
    }
    const float inv = 1.0f / red[0];

    for (int i = tid; i < TT; i += 256) scores[i] = __expf(weights[i] - gmax) * inv;

    // zero output (main kernel accumulates with atomics; harness does not re-zero)
    for (int i = tid; i < BB * DD; i += 256) out[i] = 0.0f;
}

// ---------------------------------------------------------------------------
// Kernel 2: streaming masked weighted-sum via V_WMMA_F32_16X16X4_F32.
//   grid = (DD/128, BB, TT/TCHUNK), block = 256 (8 wave32s)
//   wave w of block (x,b,z) owns d-columns [x*128 + w*16, +16) for batch b,
//   t-rows [z*TCHUNK, +TCHUNK). Accumulator D is 16x16; all 16 M-rows are
//   identical (A rows broadcast), row M=0 (c[0], lanes 0..15) is atomically
//   added into out[b, dtile].
// ---------------------------------------------------------------------------
__global__ void attn_main_kernel(const float* __restrict__ input,
                                 const long long* __restrict__ lengths,
                                 const float* __restrict__ scores,
                                 float* __restrict__ out) {
    __shared__ float cf[TCHUNK];

    const int b  = blockIdx.y;
    const int t0 = blockIdx.z * TCHUNK;
    const long long len = lengths[b];

    // Stage masked coefficients for this t-chunk into LDS (shared by all waves).
    {
        const int i = threadIdx.x;
        if (i < TCHUNK) {
            const int t = t0 + i;
            cf[i] = ((long long)t < len) ? scores[t] : 0.0f;
        }
    }
    __syncthreads();

    const int lane  = threadIdx.x & 31;
    const int wave  = threadIdx.x >> 5;
    const int dbase = blockIdx.x * 128 + wave * 16;
    const int col   = lane & 15;
    const int klo   = (lane >= 16) ? 2 : 0;   // ISA A/B layout: upper half-wave holds K=2,3

    // global pointer to input[b, t0+klo, dbase+col]
    const float* gp = input + ((size_t)b * TT + (size_t)(t0 + klo)) * DD + (size_t)(dbase + col);

    v8f c = {};  // 16x16 f32 accumulator

#pragma unroll 4
    for (int k = 0; k < TCHUNK; k += 4) {
        // A tile 16x4: row-broadcast coefficients; lane<16 -> K=0,1 ; lane>=16 -> K=2,3
        v2f a;
        a.x = cf[k + klo];
        a.y = cf[k + klo + 1];

        // B tile 4x16: 4 consecutive t-rows, 16 d-columns, streamed non-temporally
        v2f bv;
        bv.x = __builtin_nontemporal_load(gp);          // t = t0+k+klo
        bv.y = __builtin_nontemporal_load(gp + DD);     // t = t0+k+klo+1

        // prefetch 8 t-rows ahead (gfx1250 global_prefetch_b8)
        __builtin_prefetch(gp + 8 * DD, 0, 1);

        // D += A x B  (emits v_wmma_f32_16x16x4_f32)
        c = __builtin_amdgcn_wmma_f32_16x16x4_f32(
                /*neg_a=*/false, a, /*neg_b=*/false, bv,
                /*c_mod=*/(short)0, c, /*reuse_a=*/false, /*reuse_b=*/false);

        gp += 4 * DD;
    }

    // Row M=0 of D lives in c[0], lanes 0..15 (N = lane). Fold chunk partial.
    if (lane < 16) {
        unsafeAtomicAdd(&out[(size_t)b * DD + (size_t)(dbase + lane)], c[0]);
    }
}

// ---------------------------------------------------------------------------
// Harness entry
// ---------------------------------------------------------------------------
extern "C" void kernel_launch(void* const* d_in, const int* in_sizes, int n_in,
                              void* d_out, int out_size, void* d_ws, size_t ws_size,
                              hipStream_t stream) {
    const float*     input   = (const float*)d_in[0];       // [B,T,D] fp32
    const long long* lengths = (const long long*)d_in[1];   // [B] int64
    const float*     weights = (const float*)d_in[2];       // [T] fp32
    float*           out     = (float*)d_out;               // [B,D] fp32
    float*           scores  = (float*)d_ws;                // T floats of scratch

    attn_prep_kernel<<<dim3(1), dim3(256), 0, stream>>>(weights, scores, out);
    attn_main_kernel<<<dim3(DD / 128, BB, TT / TCHUNK), dim3(256), 0, stream>>>(
        input, lengths, scores, out);
}